// SpatialAttention_7275674600238
// MI455X (gfx1250) — compile-verified
//
#include <hip/hip_runtime.h>
#include <math.h>

// SpatialAttention fused pipeline for gfx1250 (MI455X).
// Memory-bound. Traffic plan with 192MB L2 exploitation:
//   K1: read x (268MB, RT -> warms L2 with high-c tail), write sa (4MB)
//   K2: 7x7x4 conv via V_WMMA_F32_16X16X4_F32 (+sigmoid) -> gate (1MB, tiny)
//   K3: read x DESCENDING in c (hits K1's L2-resident tail; NT loads),
//       write out with NT stores (don't evict x from L2).
// HBM traffic ~615MB -> ~26us floor @ 23.3 TB/s (vs 805MB/35us naive).

typedef __attribute__((ext_vector_type(4))) float f4;
typedef __attribute__((ext_vector_type(2))) float v2f;
typedef __attribute__((ext_vector_type(8))) float v8f;

#define NB 16
#define NC 256
#define HH 128
#define WW 128
#define HW (HH * WW)   // 16384
#define HW4 (HW / 4)   // 4096

// ---------------- Kernel 1: per-pixel channel stats -------------------------
// One thread per float4 group of pixels; loop c = 0..255 ascending. The last
// ~192MB touched (high-c slabs for all pixels) stays resident in L2 for K3.
__global__ __launch_bounds__(256) void stats_kernel(const f4* __restrict__ x4,
                                                    f4* __restrict__ sa4) {
    int t = blockIdx.x * 256 + threadIdx.x;   // 0 .. NB*HW4-1
    int b = t >> 12;                          // HW4 == 4096
    int q = t & (HW4 - 1);
    int base = b * NC * HW4 + q;

    f4 v = x4[base];
    f4 s = v, mx = v, mn = v;
#pragma unroll 16
    for (int c = 1; c < NC; ++c) {
        f4 u = x4[base + c * HW4];            // RT: warm L2 for K3
        s += u;
        mx.x = fmaxf(mx.x, u.x); mx.y = fmaxf(mx.y, u.y);
        mx.z = fmaxf(mx.z, u.z); mx.w = fmaxf(mx.w, u.w);
        mn.x = fminf(mn.x, u.x); mn.y = fminf(mn.y, u.y);
        mn.z = fminf(mn.z, u.z); mn.w = fminf(mn.w, u.w);
    }
    f4 mean = s * (1.0f / (float)NC);

    int sb = (b * 4) * HW4 + q;   // sa layout [B,4,H,W]: mean,max,min,sum
    sa4[sb]           = mean;
    sa4[sb + HW4]     = mx;
    sa4[sb + 2 * HW4] = mn;
    sa4[sb + 3 * HW4] = s;
}

// ---------------- Kernel 2: 7x7 conv (4->1 ch) + sigmoid via WMMA -----------
// One wave (32 lanes) per 16-pixel row segment.
//   D(16x16) += A(16x4) x B(4x16), K = the 4 stat channels, one WMMA per
//   (kh,kw) tap => 49 v_wmma_f32_16x16x4_f32 per wave. A rows are identical
//   (weights broadcast over M), so every D row holds the conv result.
// A layout (ISA 7.12.2, 32-bit A 16x4): v0 = K0(lanes0-15)/K2(lanes16-31),
//   v1 = K1/K3.  B 4x16 mirrors this: v0 = row K0 / row K2, v1 = K1 / K3.
// D layout: acc[0] lanes0-15 = row M=0, N=lane -> conv for the 16 pixels.
__global__ __launch_bounds__(32) void conv_gate_kernel(const float* __restrict__ sa,
                                                       const float* __restrict__ w,
                                                       float* __restrict__ gate) {
    __shared__ float lw[4 * 49];        // weights [c][kh][kw]
    __shared__ float lt[4 * 7 * 22];    // sa halo tile [c][7 rows][22 cols]

    int lane = threadIdx.x;             // 0..31
    int tile = blockIdx.x;              // NB*HH*(WW/16) = 16384 tiles
    int x0 = (tile & 7) << 4;           // 8 segments of 16 pixels per row
    int y  = (tile >> 3) & (HH - 1);
    int b  = tile >> 10;                // 1024 tiles per image

    for (int i = lane; i < 196; i += 32) lw[i] = w[i];

    for (int i = lane; i < 616; i += 32) {      // 4*7*22
        int c   = i / 154;
        int rem = i - c * 154;
        int r   = rem / 22;
        int col = rem - r * 22;
        int yy = y + r - 3;
        int xx = x0 + col - 3;
        float v = 0.0f;                          // zero padding
        if (yy >= 0 && yy < HH && xx >= 0 && xx < WW)
            v = sa[(b * 4 + c) * HW + yy * WW + xx];
        lt[i] = v;
    }
    __syncthreads();

    int n  = lane & 15;                  // output pixel within segment
    int c0 = (lane < 16) ? 0 : 2;        // K-pair held by this half-wave

    v8f acc = {0.0f, 0.0f, 0.0f, 0.0f, 0.0f, 0.0f, 0.0f, 0.0f};
    for (int kh = 0; kh < 7; ++kh) {
        for (int kw = 0; kw < 7; ++kw) {
            v2f a, bm;
            a.x  = lw[c0 * 49 + kh * 7 + kw];            // A[m][K=c0]   (all m)
            a.y  = lw[(c0 + 1) * 49 + kh * 7 + kw];      // A[m][K=c0+1]
            bm.x = lt[c0 * 154 + kh * 22 + n + kw];      // B[K=c0][n]
            bm.y = lt[(c0 + 1) * 154 + kh * 22 + n + kw];
            acc = __builtin_amdgcn_wmma_f32_16x16x4_f32(
                false, a, false, bm, (short)0, acc, false, false);
        }
    }

    if (lane < 16) {
        float conv = acc[0];                         // row M=0, col N=n
        float g = 1.0f / (1.0f + __expf(-conv));
        gate[b * HW + y * WW + x0 + n] = g;
    }
}

// ---------------- Kernel 3: out = x * gate ----------------------------------
// Channels walked DESCENDING so the first reads hit the x-tail K1 left in L2.
// x loads are NT (miss-fills don't evict the still-unread resident region);
// out stores are NT (write stream must not displace x lines in L2).
__global__ __launch_bounds__(256) void apply_kernel(const f4* __restrict__ x4,
                                                    const f4* __restrict__ gate4,
                                                    f4* __restrict__ out4) {
    int t = blockIdx.x * 256 + threadIdx.x;
    int b = t >> 12;
    int q = t & (HW4 - 1);
    f4 g = gate4[b * HW4 + q];           // one gate per pixel, reused over C
    int base = b * NC * HW4 + q;
#pragma unroll 8
    for (int c = NC - 1; c >= 0; --c) {
        f4 u = __builtin_nontemporal_load(&x4[base + c * HW4]);
        __builtin_nontemporal_store(u * g, &out4[base + c * HW4]);
    }
}

extern "C" void kernel_launch(void* const* d_in, const int* in_sizes, int n_in,
                              void* d_out, int out_size, void* d_ws, size_t ws_size,
                              hipStream_t stream) {
    const float* x = (const float*)d_in[0];   // [16,256,128,128]
    const float* w = (const float*)d_in[1];   // [1,4,7,7]
    float* out = (float*)d_out;               // [16,256,128,128]

    // scratch: sa = 16*4*16384 floats (4 MiB), gate = 16*16384 floats (1 MiB)
    float* sa   = (float*)d_ws;
    float* gate = sa + (size_t)NB * 4 * HW;

    (void)in_sizes; (void)n_in; (void)out_size; (void)ws_size;

    stats_kernel<<<NB * HW4 / 256, 256, 0, stream>>>((const f4*)x, (f4*)sa);
    conv_gate_kernel<<<NB * HH * (WW / 16), 32, 0, stream>>>(sa, w, gate);
    apply_kernel<<<NB * HW4 / 256, 256, 0, stream>>>((const f4*)x,
                                                     (const f4*)gate, (f4*)out);
}